// NetSeGraph_46789373722783
// MI455X (gfx1250) — compile-verified
//
#include <hip/hip_runtime.h>
#include <hip/hip_bf16.h>

// ---------------------------------------------------------------------------
// NetSeGraph on MI455X (gfx1250, wave32).
//
// All dense GEMMs run on v_wmma_f32_16x16x4_f32 (full fp32 precision; the
// workload is L2-resident and bandwidth bound, so fp32 matrix pipes are the
// right choice). Graph scatter/aggregate runs in LDS (ds_add_f32), top-k is
// an in-LDS bitonic sort (up to 1024 64-bit keys), exploiting the 320KB
// WGP LDS. The NNConv dynamic-weight contraction is refactored into two
// WMMA GEMMs via Z[e, k*128+i] = attr[e,k]*feat[src[e],i]; Wnn flat is
// already the required [2048,128] row-major B matrix.
//
// Workspace requirement: ~160 MB (see layout in kernel_launch).
// ---------------------------------------------------------------------------

#define Gc    256
#define Nc    1024
#define Ec    4096
#define K1c   512
#define K2c   256
#define K3c   128
#define Fc    64      // F_IN == NHID
#define DNc   128     // DDI_NHID (== 2*NHID)
#define EFc   16
#define EDDIc 4096

typedef __attribute__((ext_vector_type(2))) float v2f;
typedef __attribute__((ext_vector_type(8))) float v8f;

// ---------------------------------------------------------------------------
// Generic fp32 WMMA GEMM: C[M,N] = op( A'[M,K] @ B[K,N] + bias )  (+= if accum)
// A'[r,:] = A[gatherA ? gatherA[r] : r, :].   M%16==0, N%16==0, K%4==0.
// One 16x16 output tile per wave; 8 waves / 256-thread block.
// ---------------------------------------------------------------------------
__global__ __launch_bounds__(256) void k_gemm_wmma(
    const float* __restrict__ A, const float* __restrict__ B,
    const float* __restrict__ bias, float* __restrict__ C,
    const int* __restrict__ gatherA,
    int M, int N, int K, int lda, int ldb, int ldc,
    int accum, int relu)
{
    const int wave = threadIdx.x >> 5;
    const int lane = threadIdx.x & 31;
    const int tilesN = N >> 4;
    const int tiles  = (M >> 4) * tilesN;
    const int tile   = blockIdx.x * 8 + wave;
    if (tile >= tiles) return;                    // wave-uniform: EXEC stays all-ones

    const int tm   = tile / tilesN;
    const int tn   = tile - tm * tilesN;
    const int l15  = lane & 15;
    const int half = lane >> 4;
    const int rowL = tm * 16 + l15;               // logical A row for this lane
    const int col  = tn * 16 + l15;               // C/B column for this lane

    const long arow = gatherA ? (long)gatherA[rowL] : (long)rowL;
    const float* Ap = A + arow * (long)lda + 2 * half;          // K index = kb + 2*half + v
    const float* Bp = B + (long)(2 * half) * ldb + col;         // K index = kb + 2*half + v

    v8f c;
#pragma unroll
    for (int v = 0; v < 8; ++v)
        c[v] = accum ? C[(long)(tm * 16 + v + 8 * half) * ldc + col] : 0.0f;

    for (int kb = 0; kb < K; kb += 4) {
        __builtin_prefetch(Ap + kb + 16, 0, 0);   // global_prefetch on the A stream
        v2f a, b;
        a.x = Ap[kb];
        a.y = Ap[kb + 1];
        b.x = Bp[(long)kb * ldb];
        b.y = Bp[(long)(kb + 1) * ldb];
        // D = A(16x4) * B(4x16) + C   (fp32 WMMA)
        c = __builtin_amdgcn_wmma_f32_16x16x4_f32(false, a, false, b,
                                                  (short)0, c, false, false);
    }

#pragma unroll
    for (int v = 0; v < 8; ++v) {
        float val = c[v];
        if (bias) val += bias[col];
        if (relu) val = fmaxf(val, 0.0f);
        C[(long)(tm * 16 + v + 8 * half) * ldc + col] = val;
    }
}

// ---------------------------------------------------------------------------
// GCNConv scatter (per graph): self loops, symmetric rsqrt degree norm,
// scatter-add of norm * H[src] into LDS accumulator, + bias, relu.
// LDS: deg[n] + acc[n*64]  (layer1: 260KB -- needs CDNA5 WGP LDS)
// ---------------------------------------------------------------------------
__global__ __launch_bounds__(256) void k_gcn_scatter(
    const float* __restrict__ Hin,
    const int* __restrict__ srcB, const int* __restrict__ tgtB,
    const float* __restrict__ wB, int sIdx, int sW,
    const float* __restrict__ bias, float* __restrict__ Hout,
    int n, int E)
{
    const int g = blockIdx.x, tid = threadIdx.x;
    extern __shared__ float smA[];
    float* deg = smA;          // n
    float* acc = smA + n;      // n*64

    const int*   src = srcB + (long)g * sIdx;
    const int*   tgt = tgtB + (long)g * sIdx;
    const float* w   = wB   + (long)g * sW;
    const float* H   = Hin  + (long)g * n * Fc;
    float*       O   = Hout + (long)g * n * Fc;

    for (int i = tid; i < n; i += 256) deg[i] = 1.0f;          // self-loop weight
    __syncthreads();
    for (int e = tid; e < E; e += 256) atomicAdd(&deg[tgt[e]], w[e]);
    __syncthreads();
    for (int i = tid; i < n; i += 256) deg[i] = rsqrtf(deg[i]); // deg >= 1 always
    __syncthreads();
    for (int idx = tid; idx < n * Fc; idx += 256) {            // self-loop message
        const float di = deg[idx >> 6];
        acc[idx] = H[idx] * di * di;
    }
    __syncthreads();
    for (int e = tid; e < E; e += 256) {
        const int s = src[e], t = tgt[e];
        const float nrm = deg[s] * w[e] * deg[t];
        if (nrm != 0.0f) {
            const float* hs = H + s * Fc;
            float* at = acc + t * Fc;
#pragma unroll 4
            for (int f = 0; f < Fc; ++f) atomicAdd(&at[f], nrm * hs[f]);
        }
    }
    __syncthreads();
    for (int idx = tid; idx < n * Fc; idx += 256)
        O[idx] = fmaxf(acc[idx] + bias[idx & (Fc - 1)], 0.0f);
}

// sortable key: descending score, ties broken by ascending node index
__device__ __forceinline__ unsigned int fkey(float f) {
    unsigned int u = __float_as_uint(f);
    return (u & 0x80000000u) ? ~u : (u | 0x80000000u);
}

// ---------------------------------------------------------------------------
// SAGPool (per graph): GraphConv scorer, exact top-k via bitonic sort of
// 64-bit (score,index) keys in LDS, tanh gating, edge remap, fused
// max||mean readout accumulated into feat.
// ---------------------------------------------------------------------------
__global__ __launch_bounds__(256) void k_sag_pool(
    const float* __restrict__ Hin,
    const int* __restrict__ srcB, const int* __restrict__ tgtB,
    const float* __restrict__ wB, int sIdx, int sW,
    const float* __restrict__ s_root, const float* __restrict__ s_nbr,
    const float* __restrict__ s_b,
    float* __restrict__ Xout, int n, int k,
    int* __restrict__ osrc, int* __restrict__ otgt, float* __restrict__ ow,
    float* __restrict__ feat, int E)
{
    const int g = blockIdx.x, tid = threadIdx.x;
    extern __shared__ unsigned char smB[];
    float* p     = (float*)smB;                                  // n
    float* score = p + n;                                        // n
    unsigned long long* keys = (unsigned long long*)(score + n); // n (8B aligned)
    int*   newid = (int*)(keys + n);                             // n
    int*   perm  = newid + n;                                    // k
    float* gate  = (float*)(perm + k);                           // k
    float* xk    = gate + k;                                     // k*64

    const int*   src = srcB + (long)g * sIdx;
    const int*   tgt = tgtB + (long)g * sIdx;
    const float* w   = wB   + (long)g * sW;
    const float* H   = Hin  + (long)g * n * Fc;

    // score_i = h_i . s_root + b ;  p_i = h_i . s_nbr
    for (int i = tid; i < n; i += 256) {
        const float* h = H + i * Fc;
        float q = 0.f, pp = 0.f;
        for (int f = 0; f < Fc; ++f) { const float hv = h[f]; q += hv * s_root[f]; pp += hv * s_nbr[f]; }
        score[i] = q + s_b[0];
        p[i] = pp;
    }
    __syncthreads();
    for (int e = tid; e < E; e += 256)
        atomicAdd(&score[tgt[e]], w[e] * p[src[e]]);
    __syncthreads();
    for (int i = tid; i < n; i += 256)
        keys[i] = ((unsigned long long)fkey(score[i]) << 32)
                | (unsigned long long)(0xFFFFFFFFu - (unsigned)i);
    __syncthreads();
    // bitonic sort, descending (n is a power of two: 1024/512/256)
    for (int k2 = 2; k2 <= n; k2 <<= 1)
        for (int j = k2 >> 1; j > 0; j >>= 1) {
            for (int i = tid; i < n; i += 256) {
                const int ixj = i ^ j;
                if (ixj > i) {
                    const unsigned long long a = keys[i], b = keys[ixj];
                    const bool desc = ((i & k2) == 0);
                    if ((a < b) == desc) { keys[i] = b; keys[ixj] = a; }
                }
            }
            __syncthreads();
        }
    for (int j = tid; j < k; j += 256) {
        const int pi = (int)(0xFFFFFFFFu - (unsigned)(keys[j] & 0xFFFFFFFFull));
        perm[j] = pi;
        gate[j] = tanhf(score[pi]);            // top-k value = score[perm[j]]
    }
    for (int i = tid; i < n; i += 256) newid[i] = -1;
    __syncthreads();
    for (int idx = tid; idx < k * Fc; idx += 256) {
        const int j = idx >> 6, f = idx & (Fc - 1);
        const float v = H[perm[j] * Fc + f] * gate[j];
        xk[idx] = v;
        Xout[(long)g * k * Fc + idx] = v;
    }
    for (int j = tid; j < k; j += 256) newid[perm[j]] = j;
    __syncthreads();
    for (int e = tid; e < E; e += 256) {
        const int s2 = newid[src[e]], t2 = newid[tgt[e]];
        const bool valid = (s2 >= 0) && (t2 >= 0);
        osrc[(long)g * E + e] = valid ? s2 : 0;
        otgt[(long)g * E + e] = valid ? t2 : 0;
        ow  [(long)g * E + e] = valid ? w[e] : 0.0f;
    }
    // readout: concat(max, mean) over pooled nodes, accumulated (r1+r2+r3)
    if (tid < Fc) {
        float mx = -3.402823466e38f, sm = 0.f;
        for (int j = 0; j < k; ++j) { const float v = xk[j * Fc + tid]; mx = fmaxf(mx, v); sm += v; }
        feat[g * 2 * Fc + tid]      += mx;
        feat[g * 2 * Fc + Fc + tid] += sm / (float)k;
    }
}

// Z[e, k*128+i] = attr[e,k] * feat[src[e], i]
__global__ __launch_bounds__(256) void k_build_z(
    const float* __restrict__ attr, const float* __restrict__ feat,
    const int* __restrict__ src, float* __restrict__ Z)
{
    const int idx = blockIdx.x * 256 + threadIdx.x;
    if (idx >= EDDIc * (EFc * DNc)) return;
    const int e = idx >> 11, r = idx & 2047;
    const int k = r >> 7, i = r & 127;
    Z[idx] = attr[e * EFc + k] * feat[src[e] * DNc + i];
}

__global__ __launch_bounds__(256) void k_scatter_msg(
    const float* __restrict__ msg, const int* __restrict__ tgt,
    float* __restrict__ hpre)
{
    const int idx = blockIdx.x * 256 + threadIdx.x;
    if (idx >= EDDIc * DNc) return;
    const int e = idx >> 7, o = idx & 127;
    atomicAdd(&hpre[tgt[e] * DNc + o], msg[idx]);
}

__global__ __launch_bounds__(256) void k_relu(float* __restrict__ p, int n)
{
    const int i = blockIdx.x * 256 + threadIdx.x;
    if (i < n) p[i] = fmaxf(p[i], 0.0f);
}

__global__ __launch_bounds__(256) void k_zero(float* __restrict__ p, int n)
{
    const int i = blockIdx.x * 256 + threadIdx.x;
    if (i < n) p[i] = 0.0f;
}

__global__ __launch_bounds__(256) void k_rowdot(
    const float* __restrict__ X, const float* __restrict__ Y,
    float* __restrict__ out, int n)
{
    const int e = blockIdx.x * 256 + threadIdx.x;
    if (e >= n) return;
    float s = 0.f;
    for (int f = 0; f < DNc; ++f) s += X[e * DNc + f] * Y[e * DNc + f];
    out[e] = s;
}

__device__ __forceinline__ float softplus_stable(float v) {
    return fmaxf(v, 0.0f) + log1pf(expf(-fabsf(v)));
}

__global__ __launch_bounds__(256) void k_loss(
    const float* __restrict__ npos, const float* __restrict__ nneg,
    float* __restrict__ out)
{
    __shared__ float red[256];
    float s = 0.f;
    for (int e = threadIdx.x; e < EDDIc; e += 256)
        s += softplus_stable(-npos[e]) + softplus_stable(nneg[e]);
    red[threadIdx.x] = s;
    __syncthreads();
    for (int st = 128; st > 0; st >>= 1) {
        if (threadIdx.x < st) red[threadIdx.x] += red[threadIdx.x + st];
        __syncthreads();
    }
    if (threadIdx.x == 0) out[0] = red[0] / (float)EDDIc;
}

// ---------------------------------------------------------------------------
extern "C" void kernel_launch(void* const* d_in, const int* in_sizes, int n_in,
                              void* d_out, int out_size, void* d_ws, size_t ws_size,
                              hipStream_t stream)
{
    (void)in_sizes; (void)n_in; (void)out_size; (void)ws_size;

    const float* x       = (const float*)d_in[0];
    const float* ew      = (const float*)d_in[1];
    const float* dattr   = (const float*)d_in[2];
    /* d_in[3] = neg_edge_attr: unused by reference */
    const float* W1 = (const float*)d_in[4];  const float* b1 = (const float*)d_in[5];
    const float* s1r = (const float*)d_in[6]; const float* s1n = (const float*)d_in[7]; const float* s1b = (const float*)d_in[8];
    const float* W2 = (const float*)d_in[9];  const float* b2 = (const float*)d_in[10];
    const float* s2r = (const float*)d_in[11]; const float* s2n = (const float*)d_in[12]; const float* s2b = (const float*)d_in[13];
    const float* W3 = (const float*)d_in[14]; const float* b3 = (const float*)d_in[15];
    const float* s3r = (const float*)d_in[16]; const float* s3n = (const float*)d_in[17]; const float* s3b = (const float*)d_in[18];
    const float* Wnn  = (const float*)d_in[19]; const float* bnn = (const float*)d_in[20];
    const float* Wroot = (const float*)d_in[21]; const float* bc4 = (const float*)d_in[22];
    const float* Wl1 = (const float*)d_in[23]; const float* bl1 = (const float*)d_in[24];
    const float* Wl2 = (const float*)d_in[25]; const float* bl2 = (const float*)d_in[26];
    const int* ei      = (const int*)d_in[27];   // [G, 2, E]
    const int* ddi_ei  = (const int*)d_in[28];   // [2, EDDI]
    const int* neg_ei  = (const int*)d_in[29];   // [2, EDDI]

    float* out = (float*)d_out;

    // ---- workspace layout (floats). Total ~39.9M floats (~153 MiB) ----
    float* ws  = (float*)d_ws;
    float* A0  = ws;                     // 16,777,216 : GEMM out / pooled X (reused)
    float* B0  = ws + 16777216;          // 16,777,216 : GCN out / Z / msg / pos_y.. (reused)
    int*   eSrcA = (int*)(ws + 2 * 16777216);
    int*   eTgtA = eSrcA + Gc * Ec;
    float* eWA   = (float*)(eTgtA + Gc * Ec);
    int*   eSrcB = (int*)(eWA + Gc * Ec);
    int*   eTgtB = eSrcB + Gc * Ec;
    float* eWB   = (float*)(eTgtB + Gc * Ec);
    float* feat  = eWB + Gc * Ec;        // G*128
    float* hpre  = feat + Gc * 2 * Fc;   // G*128

    const int* ddi_src = ddi_ei;
    const int* ddi_tgt = ddi_ei + EDDIc;
    const int* neg_src = neg_ei;
    const int* neg_tgt = neg_ei + EDDIc;

    auto gemm = [&](const float* A, const float* B, const float* bias, float* C,
                    const int* gather, int M, int N, int K,
                    int lda, int ldb, int ldc, int accum, int relu) {
        const int tiles = (M / 16) * (N / 16);
        k_gemm_wmma<<<(tiles + 7) / 8, 256, 0, stream>>>(
            A, B, bias, C, gather, M, N, K, lda, ldb, ldc, accum, relu);
    };

    // dynamic-LDS sizes
    auto scat_lds = [](int n) { return (size_t)(n + n * Fc) * sizeof(float); };
    auto pool_lds = [](int n, int k) {
        return (size_t)n * (4 + 4 + 8 + 4) + (size_t)k * (4 + 4) + (size_t)k * Fc * 4;
    };

    k_zero<<<(Gc * 2 * Fc + 255) / 256, 256, 0, stream>>>(feat, Gc * 2 * Fc);

    // ---------------- per-graph pipeline ----------------
    // Layer 1 (n=1024 -> k=512); input edges live in d_in ([G,2,E] + weights)
    gemm(x, W1, nullptr, A0, nullptr, Gc * Nc, Fc, Fc, Fc, Fc, Fc, 0, 0);
    k_gcn_scatter<<<Gc, 256, scat_lds(Nc), stream>>>(
        A0, ei, ei + Ec, ew, 2 * Ec, Ec, b1, B0, Nc, Ec);
    k_sag_pool<<<Gc, 256, pool_lds(Nc, K1c), stream>>>(
        B0, ei, ei + Ec, ew, 2 * Ec, Ec, s1r, s1n, s1b,
        A0, Nc, K1c, eSrcA, eTgtA, eWA, feat, Ec);

    // Layer 2 (n=512 -> k=256)
    gemm(A0, W2, nullptr, A0 + 8388608, nullptr, Gc * K1c, Fc, Fc, Fc, Fc, Fc, 0, 0);
    k_gcn_scatter<<<Gc, 256, scat_lds(K1c), stream>>>(
        A0 + 8388608, eSrcA, eTgtA, eWA, Ec, Ec, b2, B0, K1c, Ec);
    k_sag_pool<<<Gc, 256, pool_lds(K1c, K2c), stream>>>(
        B0, eSrcA, eTgtA, eWA, Ec, Ec, s2r, s2n, s2b,
        A0, K1c, K2c, eSrcB, eTgtB, eWB, feat, Ec);

    // Layer 3 (n=256 -> k=128)
    gemm(A0, W3, nullptr, A0 + 8388608, nullptr, Gc * K2c, Fc, Fc, Fc, Fc, Fc, 0, 0);
    k_gcn_scatter<<<Gc, 256, scat_lds(K2c), stream>>>(
        A0 + 8388608, eSrcB, eTgtB, eWB, Ec, Ec, b3, B0, K2c, Ec);
    k_sag_pool<<<Gc, 256, pool_lds(K2c, K3c), stream>>>(
        B0, eSrcB, eTgtB, eWB, Ec, Ec, s3r, s3n, s3b,
        A0, K2c, K3c, eSrcA, eTgtA, eWA, feat, Ec);

    // ---------------- DDI head ----------------
    // msg = Z @ Wnn(as [2048,128]) + feat[src] @ bnn(as [128,128])
    float* Z    = B0;                       // [4096, 2048]
    float* msg  = B0 + 8388608;             // [4096, 128]
    float* posy = B0 + 8912896;             // [4096, 128]
    float* negx = posy + EDDIc * DNc;
    float* negy = negx + EDDIc * DNc;

    k_build_z<<<(EDDIc * EFc * DNc + 255) / 256, 256, 0, stream>>>(dattr, feat, ddi_src, Z);
    gemm(Z, Wnn, nullptr, msg, nullptr, EDDIc, DNc, EFc * DNc, EFc * DNc, DNc, DNc, 0, 0);
    gemm(feat, bnn, nullptr, msg, ddi_src, EDDIc, DNc, DNc, DNc, DNc, DNc, 1, 0);

    // hpre = feat @ Wroot + bc4 ; += segment_sum(msg, tgt) ; relu
    gemm(feat, Wroot, bc4, hpre, nullptr, Gc, DNc, DNc, DNc, DNc, DNc, 0, 0);
    k_scatter_msg<<<(EDDIc * DNc + 255) / 256, 256, 0, stream>>>(msg, ddi_tgt, hpre);
    k_relu<<<(Gc * DNc + 255) / 256, 256, 0, stream>>>(hpre, Gc * DNc);

    // link prediction heads (pos_x goes straight into d_out)
    float* posx = out + 1 + 2 * EDDIc;
    gemm(hpre, Wl1, bl1, posx, ddi_src, EDDIc, DNc, DNc, DNc, DNc, DNc, 0, 0);
    gemm(hpre, Wl2, bl2, posy, ddi_tgt, EDDIc, DNc, DNc, DNc, DNc, DNc, 0, 0);
    gemm(hpre, Wl1, bl1, negx, neg_src, EDDIc, DNc, DNc, DNc, DNc, DNc, 0, 0);
    gemm(hpre, Wl2, bl2, negy, neg_tgt, EDDIc, DNc, DNc, DNc, DNc, DNc, 0, 0);

    k_rowdot<<<(EDDIc + 255) / 256, 256, 0, stream>>>(posx, posy, out + 1, EDDIc);
    k_rowdot<<<(EDDIc + 255) / 256, 256, 0, stream>>>(negx, negy, out + 1 + EDDIc, EDDIc);
    k_loss<<<1, 256, 0, stream>>>(out + 1, out + 1 + EDDIc, out);
}